// GraphConvolutionalNetwork_29317446762855
// MI455X (gfx1250) — compile-verified
//
#include <hip/hip_runtime.h>

typedef __attribute__((ext_vector_type(2))) float v2f;
typedef __attribute__((ext_vector_type(4))) float v4f;
typedef __attribute__((ext_vector_type(8))) float v8f;

#if defined(__has_builtin)
#if __has_builtin(__builtin_amdgcn_wmma_f32_16x16x4_f32)
#define HAS_WMMA_F32X4 1
#endif
#endif
#ifndef HAS_WMMA_F32X4
#define HAS_WMMA_F32X4 0
#endif

// ---------------- utility ----------------
__global__ void k_zero(float* __restrict__ p, long long n) {
  long long i = (long long)blockIdx.x * blockDim.x + threadIdx.x;
  long long stride = (long long)gridDim.x * blockDim.x;
  for (; i < n; i += stride) p[i] = 0.0f;
}

// deg[col[e]] += ew[e]
__global__ void k_deg(const int* __restrict__ col, const float* __restrict__ ew,
                      float* __restrict__ deg, int nE) {
  int e = blockIdx.x * blockDim.x + threadIdx.x;
  if (e < nE) atomicAdd(&deg[col[e]], ew[e]);
}

// dinv = rsqrt(deg + 2); selfscale = 2*dinv*dinv  (in-place on deg buffer)
__global__ void k_dinv(float* __restrict__ dinv, float* __restrict__ selfs, int n) {
  int i = blockIdx.x * blockDim.x + threadIdx.x;
  if (i < n) {
    float d = dinv[i] + 2.0f;
    float di = rsqrtf(d);
    dinv[i] = di;
    selfs[i] = 2.0f * di * di;
  }
}

// norm[e] = dinv[row]*ew*dinv[col]   (graph-constant, reused by all 3 convs)
__global__ void k_norm(const int* __restrict__ row, const int* __restrict__ col,
                       const float* __restrict__ ew, const float* __restrict__ dinv,
                       float* __restrict__ norm, int nE) {
  int e = blockIdx.x * blockDim.x + threadIdx.x;
  if (e < nE) norm[e] = dinv[row[e]] * ew[e] * dinv[col[e]];
}

// ---------------- fp32 WMMA GEMM: C[M,N] = A[M,K] @ B[K,N] ----------------
// One wave -> one 16x16 tile. M multiple of 16, N multiple of 16, K multiple of 4.
__global__ void k_gemm_wmma_f32(const float* __restrict__ A, const float* __restrict__ B,
                                float* __restrict__ C, int M, int N, int K) {
  int lane = threadIdx.x & 31;
  int wave = blockIdx.x * (blockDim.x >> 5) + (threadIdx.x >> 5);
  int nT = N >> 4;
  int mTile = wave / nT;
  int nTile = wave - mTile * nT;
  if (mTile * 16 >= M) return;  // wave-uniform; grids are exact so never taken

  int lm = lane & 15;
  int half = lane >> 4;
  int n = nTile * 16 + lm;
  v8f acc = {};

#if HAS_WMMA_F32X4
  int m = mTile * 16 + lm;
  const float* Arow = A + (long long)m * K;
  for (int k0 = 0; k0 < K; k0 += 4) {
    int ka = k0 + 2 * half;
    v2f a;
    a.x = Arow[ka];
    a.y = Arow[ka + 1];
    v2f b;
    b.x = B[(long long)ka * N + n];
    b.y = B[(long long)(ka + 1) * N + n];
    acc = __builtin_amdgcn_wmma_f32_16x16x4_f32(false, a, false, b, (short)0, acc,
                                                false, false);
  }
#else
  for (int r = 0; r < 8; ++r) {
    int mr = mTile * 16 + r + 8 * half;
    float s = 0.0f;
    for (int k = 0; k < K; ++k) s += A[(long long)mr * K + k] * B[(long long)k * N + n];
    acc[r] = s;
  }
#endif

#pragma unroll
  for (int r = 0; r < 8; ++r) {
    int mr = mTile * 16 + r + 8 * half;
    C[(long long)mr * N + n] = acc[r];
  }
}

// ------- edge scatter, 4 dims/thread: agg[col][d..d+3] += norm[e]*xw[row][d..d+3] -------
// qShift = log2(N/4). total threads = E * N/4.
__global__ void k_scatter4(const int* __restrict__ row, const int* __restrict__ col,
                           const float* __restrict__ norm, const float* __restrict__ xw,
                           float* __restrict__ agg, long long total, int nShift, int qShift) {
  long long gid = (long long)blockIdx.x * blockDim.x + threadIdx.x;
  if (gid >= total) return;
  int e = (int)(gid >> qShift);
  int d = ((int)gid & ((1 << qShift) - 1)) << 2;
  float w = norm[e];
  const v4f* src = (const v4f*)(xw + ((long long)row[e] << nShift) + d);
  v4f v = *src;
  float* dst = agg + ((long long)col[e] << nShift) + d;
  atomicAdd(dst + 0, w * v.x);
  atomicAdd(dst + 1, w * v.y);
  atomicAdd(dst + 2, w * v.z);
  atomicAdd(dst + 3, w * v.w);
}

// scalar scatter (N==1): agg[col] += norm[e]*xw[row]
__global__ void k_scatter1(const int* __restrict__ row, const int* __restrict__ col,
                           const float* __restrict__ norm, const float* __restrict__ xw,
                           float* __restrict__ agg, int nE) {
  int e = blockIdx.x * blockDim.x + threadIdx.x;
  if (e < nE) atomicAdd(&agg[col[e]], norm[e] * xw[row[e]]);
}

// agg[i][d..d+3] += selfscale[i]*xw[i][d..d+3] + bias[d..d+3]  (4 elems/thread)
__global__ void k_combine4(float* __restrict__ agg, const float* __restrict__ xw,
                           const float* __restrict__ selfs, const float* __restrict__ bias,
                           long long total4, int qShift) {
  long long gid = (long long)blockIdx.x * blockDim.x + threadIdx.x;
  if (gid >= total4) return;
  int i = (int)(gid >> qShift);
  int d = ((int)gid & ((1 << qShift) - 1)) << 2;
  float s = selfs[i];
  const v4f* bp = (const v4f*)(bias + d);
  v4f b = *bp;
  v4f* ap = (v4f*)(agg + (gid << 2));
  const v4f* xp = (const v4f*)(xw + (gid << 2));
  v4f a = *ap;
  v4f x = *xp;
  a.x += s * x.x + b.x;
  a.y += s * x.y + b.y;
  a.z += s * x.z + b.z;
  a.w += s * x.w + b.w;
  *ap = a;
}

// scalar combine (N==1)
__global__ void k_combine1(float* __restrict__ agg, const float* __restrict__ xw,
                           const float* __restrict__ selfs, const float* __restrict__ bias,
                           int n) {
  int i = blockIdx.x * blockDim.x + threadIdx.x;
  if (i < n) agg[i] += selfs[i] * xw[i] + bias[0];
}

// per-column sum and sum-of-squares -> stats[0:N], stats[N:2N]
__global__ void k_bnstats(const float* __restrict__ v, float* __restrict__ stats,
                          int M, int N) {
  __shared__ float ssum[256];
  __shared__ float ssq[256];
  int tid = threadIdx.x;
  int col = tid & (N - 1);
  int rpi = 256 / N;  // rows per block-iteration
  int r0 = tid / N;
  float s = 0.0f, q = 0.0f;
  for (long long r = (long long)blockIdx.x * rpi + r0; r < M;
       r += (long long)gridDim.x * rpi) {
    float x = v[r * N + col];
    s += x;
    q += x * x;
  }
  ssum[tid] = s;
  ssq[tid] = q;
  __syncthreads();
  if (tid < N) {
    float S = 0.0f, Q = 0.0f;
    for (int t = tid; t < 256; t += N) { S += ssum[t]; Q += ssq[t]; }
    atomicAdd(&stats[col], S);
    atomicAdd(&stats[N + col], Q);
  }
}

// out = relu((v-mu)*rsqrt(var+eps)*g + bt) [+ resid], 4 elems/thread
__global__ void k_bnapply4(const float* __restrict__ v, const float* __restrict__ stats,
                           const float* __restrict__ g, const float* __restrict__ bt,
                           const float* __restrict__ resid, float* __restrict__ out,
                           long long total4, int qShift, int N, float invM) {
  long long gid = (long long)blockIdx.x * blockDim.x + threadIdx.x;
  if (gid >= total4) return;
  int d = ((int)gid & ((1 << qShift) - 1)) << 2;
  v4f mu = *(const v4f*)(stats + d);
  v4f sq = *(const v4f*)(stats + N + d);
  v4f gg = *(const v4f*)(g + d);
  v4f bb = *(const v4f*)(bt + d);
  v4f x = *(const v4f*)(v + (gid << 2));
  float m0 = mu.x * invM, m1 = mu.y * invM, m2 = mu.z * invM, m3 = mu.w * invM;
  float i0 = rsqrtf(sq.x * invM - m0 * m0 + 1e-5f);
  float i1 = rsqrtf(sq.y * invM - m1 * m1 + 1e-5f);
  float i2 = rsqrtf(sq.z * invM - m2 * m2 + 1e-5f);
  float i3 = rsqrtf(sq.w * invM - m3 * m3 + 1e-5f);
  v4f r;
  r.x = fmaxf((x.x - m0) * i0 * gg.x + bb.x, 0.0f);
  r.y = fmaxf((x.y - m1) * i1 * gg.y + bb.y, 0.0f);
  r.z = fmaxf((x.z - m2) * i2 * gg.z + bb.z, 0.0f);
  r.w = fmaxf((x.w - m3) * i3 * gg.w + bb.w, 0.0f);
  if (resid) {
    v4f rs = *(const v4f*)(resid + (gid << 2));
    r.x += rs.x; r.y += rs.y; r.z += rs.z; r.w += rs.w;
  }
  *(v4f*)(out + (gid << 2)) = r;
}

// out[i] = dot(h[i, 0:32], W[0:32])
__global__ void k_matvec32(const float* __restrict__ h, const float* __restrict__ W,
                           float* __restrict__ out, int M) {
  int i = blockIdx.x * blockDim.x + threadIdx.x;
  if (i < M) {
    const float* hr = h + (long long)i * 32;
    float s = 0.0f;
#pragma unroll
    for (int k = 0; k < 32; ++k) s += hr[k] * W[k];
    out[i] = s;
  }
}

__global__ void k_reduce(const float* __restrict__ v, float* __restrict__ out, int n) {
  __shared__ float sm[256];
  int tid = threadIdx.x;
  float s = 0.0f;
  for (long long i = (long long)blockIdx.x * blockDim.x + tid; i < n;
       i += (long long)gridDim.x * blockDim.x)
    s += v[i];
  sm[tid] = s;
  __syncthreads();
  for (int off = 128; off > 0; off >>= 1) {
    if (tid < off) sm[tid] += sm[tid + off];
    __syncthreads();
  }
  if (tid == 0) atomicAdd(out, sm[0]);
}

__global__ void k_final(const float* __restrict__ pooled, const float* __restrict__ fw1,
                        const float* __restrict__ fb1, const float* __restrict__ fg,
                        const float* __restrict__ fbt, const float* __restrict__ fw2,
                        const float* __restrict__ fb2, float* __restrict__ out, float invM) {
  if (threadIdx.x == 0 && blockIdx.x == 0) {
    float p = pooled[0] * invM;
    float y = p * fw1[0] + fb1[0];
    y = y * rsqrtf(1.0f + 1e-5f) * fg[0] + fbt[0];
    y = fmaxf(y, 0.0f);
    y = y * fw2[0] + fb2[0];
    out[0] = y;
  }
}

static inline int nblk(long long n, int b) { return (int)((n + b - 1) / b); }

extern "C" void kernel_launch(void* const* d_in, const int* in_sizes, int n_in,
                              void* d_out, int out_size, void* d_ws, size_t ws_size,
                              hipStream_t stream) {
  const float* x   = (const float*)d_in[0];
  const int*   ei  = (const int*)d_in[1];
  const float* ew  = (const float*)d_in[2];
  const float* W1  = (const float*)d_in[3];
  const float* b1  = (const float*)d_in[4];
  const float* g1  = (const float*)d_in[5];
  const float* bt1 = (const float*)d_in[6];
  const float* W2  = (const float*)d_in[7];
  const float* b2  = (const float*)d_in[8];
  const float* g2  = (const float*)d_in[9];
  const float* bt2 = (const float*)d_in[10];
  const float* W3  = (const float*)d_in[11];
  const float* b3  = (const float*)d_in[12];
  const float* fw1 = (const float*)d_in[13];
  const float* fb1 = (const float*)d_in[14];
  const float* fg  = (const float*)d_in[15];
  const float* fbt = (const float*)d_in[16];
  const float* fw2 = (const float*)d_in[17];
  const float* fb2 = (const float*)d_in[18];
  float* out = (float*)d_out;

  const int M = in_sizes[0] / 64;  // 100000 nodes
  const int E = in_sizes[2];       // 1600000 edges
  const int* row  = ei;
  const int* colv = ei + E;
  const float invM = 1.0f / (float)M;

  // workspace layout (floats): ~59.2 MB total; all blocks 16B aligned
  float* ws    = (float*)d_ws;
  float* dinv  = ws;                          // M    (deg then dinv)
  float* selfs = dinv + M;                    // M
  float* norm  = selfs + M;                   // E
  float* bufA  = norm + E;                    // M*64 (xw1 -> h1 -> h2)
  float* bufB  = bufA + (long long)M * 64;    // M*64 (agg1 -> xw2|agg2)
  float* stats = bufB + (long long)M * 64;    // 256
  float* xw3   = stats + 256;                 // M
  float* agg3  = xw3 + M;                     // M
  float* pooled = agg3 + M;                   // 1

  // --- init / normalization (shared by all conv layers) ---
  k_zero<<<nblk(M, 256), 256, 0, stream>>>(dinv, M);
  k_zero<<<2048, 256, 0, stream>>>(bufB, (long long)M * 64);
  k_zero<<<1, 256, 0, stream>>>(stats, 256);
  k_zero<<<nblk(M, 256), 256, 0, stream>>>(agg3, M);
  k_zero<<<1, 32, 0, stream>>>(pooled, 1);

  k_deg<<<nblk(E, 256), 256, 0, stream>>>(colv, ew, dinv, E);
  k_dinv<<<nblk(M, 256), 256, 0, stream>>>(dinv, selfs, M);
  k_norm<<<nblk(E, 256), 256, 0, stream>>>(row, colv, ew, dinv, norm, E);

  // --- layer 1: conv(64->64) -> BN -> relu -> +x ---
  {
    long long waves = (long long)(M / 16) * 4;  // 25000, exact
    k_gemm_wmma_f32<<<nblk(waves * 32, 128), 128, 0, stream>>>(x, W1, bufA, M, 64, 64);
    long long tot = (long long)E * 16;  // E * N/4
    k_scatter4<<<nblk(tot, 256), 256, 0, stream>>>(row, colv, norm, bufA, bufB, tot, 6, 4);
    long long tn4 = (long long)M * 16;  // M * N/4
    k_combine4<<<nblk(tn4, 256), 256, 0, stream>>>(bufB, bufA, selfs, b1, tn4, 4);
    k_bnstats<<<240, 256, 0, stream>>>(bufB, stats, M, 64);
    k_bnapply4<<<nblk(tn4, 256), 256, 0, stream>>>(bufB, stats, g1, bt1, x, bufA, tn4, 4, 64, invM);
  }

  // --- layer 2: conv(64->32) -> BN -> relu ---
  {
    float* xw2  = bufB;
    float* agg2 = bufB + (long long)M * 32;
    long long waves = (long long)(M / 16) * 2;  // 12500, exact
    k_gemm_wmma_f32<<<nblk(waves * 32, 128), 128, 0, stream>>>(bufA, W2, xw2, M, 32, 64);
    k_zero<<<2048, 256, 0, stream>>>(agg2, (long long)M * 32);
    k_zero<<<1, 256, 0, stream>>>(stats, 256);
    long long tot = (long long)E * 8;  // E * N/4
    k_scatter4<<<nblk(tot, 256), 256, 0, stream>>>(row, colv, norm, xw2, agg2, tot, 5, 3);
    long long tn4 = (long long)M * 8;
    k_combine4<<<nblk(tn4, 256), 256, 0, stream>>>(agg2, xw2, selfs, b2, tn4, 3);
    k_bnstats<<<240, 256, 0, stream>>>(agg2, stats, M, 32);
    k_bnapply4<<<nblk(tn4, 256), 256, 0, stream>>>(agg2, stats, g2, bt2, nullptr, bufA, tn4, 3, 32, invM);
  }

  // --- layer 3: conv(32->1) -> mean pool -> scalar MLP ---
  {
    k_matvec32<<<nblk(M, 256), 256, 0, stream>>>(bufA, W3, xw3, M);
    k_scatter1<<<nblk(E, 256), 256, 0, stream>>>(row, colv, norm, xw3, agg3, E);
    k_combine1<<<nblk(M, 256), 256, 0, stream>>>(agg3, xw3, selfs, b3, M);
    k_reduce<<<256, 256, 0, stream>>>(agg3, pooled, M);
    k_final<<<1, 32, 0, stream>>>(pooled, fw1, fb1, fg, fbt, fw2, fb2, out, invM);
  }
}